// SPD_GBMS_RNN_20959440405156
// MI455X (gfx1250) — compile-verified
//
#include <hip/hip_runtime.h>
#include <cstddef>

// SPD Gaussian-blurred mean shift on the SPD manifold, MI455X (gfx1250).
// Eigendecomposition-free: logm via Newton-Schulz inverse scaling&squaring,
// expm via Taylor scaling&squaring -> everything is 64x64x64 f32 GEMMs on
// LDS, executed with V_WMMA_F32_16X16X4_F32 (wave32, 8 waves/block).

#define NMAT 256
#define D64  64
#define LDM  65            // padded LDS row stride (floats)
#define MSZ  (D64 * LDM)   // floats per LDS matrix buffer

typedef float v2f __attribute__((ext_vector_type(2)));
typedef float v8f __attribute__((ext_vector_type(8)));

__device__ __forceinline__ v8f wmma4(v2f a, v2f b, v8f c) {
  // D(16x16,f32) = A(16x4,f32) * B(4x16,f32) + C
  return __builtin_amdgcn_wmma_f32_16x16x4_f32(
      /*neg_a=*/false, a, /*neg_b=*/false, b,
      /*c_mod=*/(short)0, c, /*reuse_a=*/false, /*reuse_b=*/false);
}

// One wave accumulates a 16x16 tile of A(64xK=64) * B(64x64) from LDS/smem.
// A layout (row-major, stride lda):  A[(ti+m)*lda + k]
// B layout (row-major, stride ldb):  B[k*ldb + (tj+n)]
__device__ __forceinline__ void wave_mma64(const float* __restrict__ A, int lda,
                                           const float* __restrict__ B, int ldb,
                                           int ti, int tj, v8f& acc) {
  const int lane = threadIdx.x & 31;
  const int half = lane >> 4;   // 0: K pair {0,1}, 1: K pair {2,3}
  const int r    = lane & 15;   // M for A-operand, N for B-operand
#pragma unroll
  for (int kk = 0; kk < 64; kk += 4) {
    const int k0 = kk + 2 * half;
    v2f a, b;
    a.x = A[(ti + r) * lda + k0];
    a.y = A[(ti + r) * lda + k0 + 1];
    b.x = B[k0 * ldb + tj + r];
    b.y = B[(k0 + 1) * ldb + tj + r];
    acc = wmma4(a, b, acc);
  }
}

// Block-level (256 threads = 8 waves): D = alpha*(A*B) + beta*I  (64x64, LDS).
// Caller must __syncthreads() around calls.
__device__ __forceinline__ void gemm64(const float* __restrict__ A,
                                       const float* __restrict__ B,
                                       float* __restrict__ D,
                                       float alpha, float beta) {
  const int wave = threadIdx.x >> 5;
  const int lane = threadIdx.x & 31;
  const int n = lane & 15, hh = lane >> 4;
#pragma unroll
  for (int t = 0; t < 2; ++t) {
    const int tile = wave * 2 + t;            // 0..15
    const int ti = (tile >> 2) * 16, tj = (tile & 3) * 16;
    v8f acc = {};
    wave_mma64(A, LDM, B, LDM, ti, tj, acc);
#pragma unroll
    for (int rr = 0; rr < 8; ++rr) {
      const int gi = ti + rr + 8 * hh, gj = tj + n;
      D[gi * LDM + gj] = alpha * acc[rr] + ((gi == gj) ? beta : 0.0f);
    }
  }
}

__device__ __forceinline__ float block_sum(float v, float* red) {
  const int tid = threadIdx.x;
  red[tid] = v;
  __syncthreads();
  for (int s = 128; s > 0; s >>= 1) {
    if (tid < s) red[tid] += red[tid + s];
    __syncthreads();
  }
  float r = red[0];
  __syncthreads();
  return r;
}

// ---------------- Kernel 1: logm(X) per matrix --------------------------
// 5x Newton-Schulz sqrt stages (X^(1/32)), then 9-term Gregory series.
// Also emits sq[i]=||logX||_F^2 and s[i]=sum(logX).
__global__ __launch_bounds__(256) void k_logm(const float* __restrict__ X,
                                              float* __restrict__ Lout,
                                              float* __restrict__ sqOut,
                                              float* __restrict__ sOut) {
  extern __shared__ float sm[];
  float* B0 = sm;
  float* Y = sm + MSZ;
  float* Z = sm + 2 * MSZ;
  float* U = sm + 3 * MSZ;
  const int tid = threadIdx.x, mat = blockIdx.x;
  const float* Xg = X + (size_t)mat * 4096;

  float* S = B0;  // stays B0 throughout; doubles as T inside NS iterations
  for (int e = tid; e < 4096; e += 256)
    S[(e >> 6) * LDM + (e & 63)] = Xg[e];
  __syncthreads();

  for (int stage = 0; stage < 5; ++stage) {
    // Frobenius scaling so eigenvalues of c*S lie in (0,1]
    float ps = 0.f;
    for (int e = tid; e < 4096; e += 256) {
      float x = S[(e >> 6) * LDM + (e & 63)];
      ps += x * x;
    }
    const float f = sqrtf(block_sum(ps, U));
    const float c = 1.0f / f;
    for (int e = tid; e < 4096; e += 256) {
      const int i = e >> 6, j = e & 63;
      Y[i * LDM + j] = c * S[i * LDM + j];
      Z[i * LDM + j] = (i == j) ? 1.0f : 0.0f;
    }
    __syncthreads();

    float* T = S;  // S's data is already consumed into Y; reuse as T
    const int ni = (stage == 0) ? 16 : (stage == 1 ? 12 : 9);
    for (int it = 0; it < ni; ++it) {
      gemm64(Z, Y, T, -1.0f, 3.0f);  __syncthreads();  // T = 3I - Z*Y
      gemm64(Y, T, U, 0.5f, 0.0f);   __syncthreads();  // U = 0.5*Y*T  (Y')
      gemm64(T, Z, Y, 0.5f, 0.0f);   __syncthreads();  // oldY <- 0.5*T*Z (Z')
      float* oy = Y; Y = U; Z = oy; U = (U == Y) ? U : U; // rotate below
      // proper rotation: Y'=U, Z'=oldY, free=oldZ
      // (computed above; fix U:)
      float* oz = (oy == Z) ? nullptr : nullptr; (void)oz;
      U = (oy == Z) ? U : U;
      // NOTE: explicit rotation done here:
      // before: Y=oldY, Z=oldZ, U=oldU ; writes: U<-Y', oldY<-Z'
      // after : Y=oldU, Z=oldY, U=oldZ
      // the three lines above already set Y and Z; recover U = oldZ:
      // oldZ is whichever of the three buffers is neither Y nor Z now.
      {
        float* b1 = sm + MSZ; float* b2 = sm + 2 * MSZ; float* b3 = sm + 3 * MSZ;
        if (b1 != Y && b1 != Z) U = b1;
        else if (b2 != Y && b2 != Z) U = b2;
        else U = b3;
      }
    }
    // S <- sqrt(f) * Y   (sqrt(S_old)); S aliases T (B0), safe.
    const float sf = sqrtf(f);
    for (int e = tid; e < 4096; e += 256) {
      const int i = e >> 6, j = e & 63;
      S[i * LDM + j] = sf * Y[i * LDM + j];
    }
    __syncthreads();
  }

  // E = S - I ; log(I+E) via Horner, a_j = (-1)^(j+1)/j, order 9
  float* E = Y;
  for (int e = tid; e < 4096; e += 256) {
    const int i = e >> 6, j = e & 63;
    E[i * LDM + j] = S[i * LDM + j] - ((i == j) ? 1.0f : 0.0f);
  }
  float* Q = Z; float* V = U;
  for (int e = tid; e < 4096; e += 256) {
    const int i = e >> 6, j = e & 63;
    Q[i * LDM + j] = (i == j) ? (1.0f / 9.0f) : 0.0f;
  }
  __syncthreads();
  for (int j = 8; j >= 1; --j) {
    const float aj = ((j & 1) ? 1.0f : -1.0f) / (float)j;
    gemm64(E, Q, V, 1.0f, aj);  __syncthreads();  // V = E*Q + aj*I
    float* t = Q; Q = V; V = t;
  }
  gemm64(E, Q, V, 1.0f, 0.0f);  __syncthreads();  // V = log(I+E)

  // logX = 32 * V ; emit Frobenius^2 and entry-sum
  float psq = 0.f, pss = 0.f;
  for (int e = tid; e < 4096; e += 256) {
    const float v = 32.0f * V[(e >> 6) * LDM + (e & 63)];
    Lout[(size_t)mat * 4096 + e] = v;
    psq += v * v;
    pss += v;
  }
  const float tsq = block_sum(psq, E);
  const float tss = block_sum(pss, E);
  if (tid == 0) { sqOut[mat] = tsq; sOut[mat] = tss; }
}

// ---------------- Kernel 2: W = exp(-pds / (2 bw^2)) --------------------
// Gram over flattened logX (K=4096) with WMMA, fused kernel epilogue.
__global__ __launch_bounds__(256) void k_gram_w(const float* __restrict__ L,
                                                const float* __restrict__ sq,
                                                const float* __restrict__ s,
                                                const float* __restrict__ bwp,
                                                float* __restrict__ W) {
  extern __shared__ float sm[];
  float* At = sm;        // At[m][p'] = L[i0+m][p0+p']
  float* Bt = sm + MSZ;  // Bt[p'][n] = L[j0+n][p0+p']
  const int tid = threadIdx.x;
  const int i0 = blockIdx.y * 64, j0 = blockIdx.x * 64;
  const int wave = tid >> 5, lane = tid & 31, n = lane & 15, hh = lane >> 4;

  v8f acc0 = {}, acc1 = {};
  for (int p0 = 0; p0 < 4096; p0 += 64) {
    for (int e = tid; e < 4096; e += 256) {
      const int r = e >> 6, cdx = e & 63;
      At[r * LDM + cdx] = L[(size_t)(i0 + r) * 4096 + p0 + cdx];
      Bt[cdx * LDM + r] = L[(size_t)(j0 + r) * 4096 + p0 + cdx];
    }
    __syncthreads();
    int tile = wave * 2;
    wave_mma64(At, LDM, Bt, LDM, (tile >> 2) * 16, (tile & 3) * 16, acc0);
    tile = wave * 2 + 1;
    wave_mma64(At, LDM, Bt, LDM, (tile >> 2) * 16, (tile & 3) * 16, acc1);
    __syncthreads();
  }
  const float eps = 1e-7f;
  const float bw = bwp[0];
  const float inv2bw2 = 0.5f / (bw * bw);
#pragma unroll
  for (int t = 0; t < 2; ++t) {
    v8f acc = t ? acc1 : acc0;
    const int tile = wave * 2 + t;
    const int ti = (tile >> 2) * 16, tj = (tile & 3) * 16;
#pragma unroll
    for (int rr = 0; rr < 8; ++rr) {
      const int gi = i0 + ti + rr + 8 * hh;
      const int gj = j0 + tj + n;
      const float pds = sq[gi] + sq[gj] - 2.0f * acc[rr]
                      + 2.0f * eps * (s[gj] - s[gi]) + eps * eps * 4096.0f;
      W[gi * 256 + gj] = expf(-pds * inv2bw2);
    }
  }
}

// ---------------- Kernel 3: row/col sums of W ---------------------------
__global__ void k_sums(const float* __restrict__ W, float* __restrict__ rs,
                       float* __restrict__ cs) {
  const int k = threadIdx.x;
  float r = 0.f, c = 0.f;
  for (int j = 0; j < 256; ++j) {
    r += W[k * 256 + j];
    c += W[j * 256 + k];
  }
  rs[k] = r;
  cs[k] = c;
}

// ---------------- Kernel 4: G = logX + (W^T logX - colsum*logX)/rowsum --
__global__ __launch_bounds__(256) void k_shift(const float* __restrict__ W,
                                               const float* __restrict__ L,
                                               const float* __restrict__ rs,
                                               const float* __restrict__ cs,
                                               float* __restrict__ G) {
  extern __shared__ float sm[];
  float* Wt = sm;        // Wt[m][j'] = W[kk+j'][k0+m]   (A operand = W^T tile)
  float* Bt = sm + MSZ;  // Bt[j'][n] = L[kk+j'][p0+n]
  const int tid = threadIdx.x;
  const int k0 = (blockIdx.x & 3) * 64;
  const int p0 = (blockIdx.x >> 2) * 64;
  const int wave = tid >> 5, lane = tid & 31, n = lane & 15, hh = lane >> 4;

  v8f acc0 = {}, acc1 = {};
  for (int kk = 0; kk < 256; kk += 64) {
    for (int e = tid; e < 4096; e += 256) {
      const int r = e >> 6, cdx = e & 63;
      Wt[cdx * LDM + r] = W[(kk + r) * 256 + k0 + cdx];
      Bt[r * LDM + cdx] = L[(size_t)(kk + r) * 4096 + p0 + cdx];
    }
    __syncthreads();
    int tile = wave * 2;
    wave_mma64(Wt, LDM, Bt, LDM, (tile >> 2) * 16, (tile & 3) * 16, acc0);
    tile = wave * 2 + 1;
    wave_mma64(Wt, LDM, Bt, LDM, (tile >> 2) * 16, (tile & 3) * 16, acc1);
    __syncthreads();
  }
#pragma unroll
  for (int t = 0; t < 2; ++t) {
    v8f acc = t ? acc1 : acc0;
    const int tile = wave * 2 + t;
    const int ti = (tile >> 2) * 16, tj = (tile & 3) * 16;
#pragma unroll
    for (int rr = 0; rr < 8; ++rr) {
      const int kg = k0 + ti + rr + 8 * hh;
      const int pg = p0 + tj + n;
      const float lx = L[(size_t)kg * 4096 + pg];
      G[(size_t)kg * 4096 + pg] = lx + (acc[rr] - cs[kg] * lx) / rs[kg];
    }
  }
}

// ---------------- Kernel 5: out = expm(G) in place ----------------------
__global__ __launch_bounds__(256) void k_expm(float* __restrict__ GO) {
  extern __shared__ float sm[];
  float* A = sm;
  float* P = sm + MSZ;
  float* U = sm + 2 * MSZ;
  const int tid = threadIdx.x, mat = blockIdx.x;
  float* Gg = GO + (size_t)mat * 4096;

  for (int e = tid; e < 4096; e += 256)
    A[(e >> 6) * LDM + (e & 63)] = Gg[e];
  __syncthreads();

  float ps = 0.f;
  for (int e = tid; e < 4096; e += 256) {
    const float x = A[(e >> 6) * LDM + (e & 63)];
    ps += x * x;
  }
  float f = sqrtf(block_sum(ps, P));
  int nsq = 0;
  while (f > 0.30f && nsq < 12) { f *= 0.5f; ++nsq; }  // uniform per block
  const float scale = exp2f((float)(-nsq));
  for (int e = tid; e < 4096; e += 256) {
    const int i = e >> 6, j = e & 63;
    A[i * LDM + j] *= scale;
    P[i * LDM + j] = (i == j) ? 1.0f : 0.0f;
  }
  __syncthreads();
  for (int j = 12; j >= 1; --j) {       // P <- I + (A*P)/j  (Horner exp)
    gemm64(A, P, U, 1.0f / (float)j, 1.0f);  __syncthreads();
    float* t = P; P = U; U = t;
  }
  for (int q = 0; q < nsq; ++q) {       // repeated squaring
    gemm64(P, P, U, 1.0f, 0.0f);  __syncthreads();
    float* t = P; P = U; U = t;
  }
  for (int e = tid; e < 4096; e += 256)
    Gg[e] = P[(e >> 6) * LDM + (e & 63)];
}

// ------------------------------------------------------------------------
extern "C" void kernel_launch(void* const* d_in, const int* in_sizes, int n_in,
                              void* d_out, int out_size, void* d_ws,
                              size_t ws_size, hipStream_t stream) {
  (void)in_sizes; (void)n_in; (void)out_size; (void)ws_size;
  const float* X  = (const float*)d_in[0];   // [256,64,64] f32
  const float* bw = (const float*)d_in[1];   // scalar f32
  float* out = (float*)d_out;                // [256,64,64] f32 (also G scratch)

  float* wsf = (float*)d_ws;                 // needs ~4.5 MB
  float* L  = wsf;                           // 256*4096
  float* W  = L + (size_t)256 * 4096;        // 256*256
  float* rs = W + 256 * 256;                 // 256
  float* cs = rs + 256;                      // 256
  float* sq = cs + 256;                      // 256
  float* sA = sq + 256;                      // 256

  const size_t sh1 = (size_t)(4 * MSZ) * sizeof(float);  // ~66.5 KB
  const size_t sh2 = (size_t)(2 * MSZ) * sizeof(float);  // ~33.3 KB
  const size_t sh5 = (size_t)(3 * MSZ) * sizeof(float);  // ~49.9 KB

  // k_logm needs >64KB dynamic LDS (gfx1250 WGP has 320KB). Idempotent call.
  hipFuncSetAttribute(reinterpret_cast<const void*>(k_logm),
                      hipFuncAttributeMaxDynamicSharedMemorySize, (int)sh1);

  k_logm<<<NMAT, 256, sh1, stream>>>(X, L, sq, sA);
  k_gram_w<<<dim3(4, 4), 256, sh2, stream>>>(L, sq, sA, bw, W);
  k_sums<<<1, 256, 0, stream>>>(W, rs, cs);
  k_shift<<<NMAT, 256, sh2, stream>>>(W, L, rs, cs, out);
  k_expm<<<NMAT, 256, sh5, stream>>>(out);
}